// SimpleAttention_61710090109802
// MI455X (gfx1250) — compile-verified
//
#include <hip/hip_runtime.h>
#include <hip/hip_bf16.h>
#include <math.h>

typedef __attribute__((ext_vector_type(2))) float v2f;
typedef __attribute__((ext_vector_type(8))) float v8f;

#define B_  64
#define T_  1024
#define QD_ 1024
#define CD_ 512

// ---------------------------------------------------------------------------
// Phase 1: a = ((1-u)*alpha + u*shift(alpha) + 1e-6)^sq, normalized over T.
// One block per batch. Cheap (64K elements total).
// ---------------------------------------------------------------------------
__global__ __launch_bounds__(256) void alpha_kernel(
    const float* __restrict__ alpha, const float* __restrict__ u,
    const float* __restrict__ sq, float* __restrict__ a_out) {
  const int b   = blockIdx.x;
  const float ub  = u[b];
  const float sqb = sq[b];
  const float omu = 1.0f - ub;

  __shared__ float red[256];
  __shared__ float s_inv;

  float w[4];
  float partial = 0.0f;
#pragma unroll
  for (int k = 0; k < 4; ++k) {
    const int t = threadIdx.x + k * 256;
    const float cur  = alpha[b * T_ + t];
    const float prev = (t > 0) ? alpha[b * T_ + t - 1] : 0.0f;
    const float x = omu * cur + ub * prev + 1e-6f;        // > 0 always
    w[k] = __expf(sqb * __logf(x));                       // pow(x, sq)
    partial += w[k];
  }
  red[threadIdx.x] = partial;
  __syncthreads();
  for (int s = 128; s > 0; s >>= 1) {
    if (threadIdx.x < s) red[threadIdx.x] += red[threadIdx.x + s];
    __syncthreads();
  }
  if (threadIdx.x == 0) s_inv = 1.0f / red[0];
  __syncthreads();
  const float inv = s_inv;
#pragma unroll
  for (int k = 0; k < 4; ++k) {
    const int t = threadIdx.x + k * 256;
    a_out[b * T_ + t] = w[k] * inv;
  }
}

// ---------------------------------------------------------------------------
// Phase 2: context[b,c] = sum_t a[b,t] * inputs[b,t,c]   (streams 128 MB)
// One wave owns a 16-channel tile of one batch. Per step of 4 timesteps:
//   A (16x4 f32): a[t0..t0+3] replicated across all 16 M-rows.
//     Layout (ISA 7.12.2): lane = M; VGPR0 = K0 (lanes 0-15) / K2 (16-31),
//     VGPR1 = K1 / K3  ->  lane loads (a[t0+2*half], a[t0+2*half+1]).
//   B (4x16 f32): inputs[t0+k, c0+n]; same half-split over K.
// D accumulates over all 256 steps; every row of D equals the context tile.
// ---------------------------------------------------------------------------
__global__ __launch_bounds__(256) void context_kernel(
    const float* __restrict__ inputs, const float* __restrict__ a_norm,
    float* __restrict__ out_ctx) {
  const int wave = threadIdx.x >> 5;
  const int lane = threadIdx.x & 31;
  const int w    = blockIdx.x * 8 + wave;   // global wave id: 0..2047
  const int b    = w >> 5;                  // 32 tiles per batch
  const int c0   = (w & 31) << 4;           // channel tile base
  const int half = lane >> 4;               // 0: K{0,1}, 1: K{2,3}
  const int n    = lane & 15;               // column within tile

  // Stage normalized alpha for this batch in LDS (4 KB); every wave in the
  // block shares the same batch b (8 waves = 8 tiles of one batch).
  __shared__ float sa[T_];
  for (int i = threadIdx.x; i < T_; i += 256) sa[i] = a_norm[b * T_ + i];
  __syncthreads();

  // Per-lane streaming pointer: row (t + 2*half), column (c0 + n).
  const float* p = inputs + ((size_t)b * T_ + 2 * half) * CD_ + c0 + n;

  v8f acc = {};
  for (int t0 = 0; t0 < T_; t0 += 16) {
    // Prefetch 64 rows (16 steps) ahead of the stream into the caches.
    __builtin_prefetch(p + 64 * CD_, 0, 0);
#pragma unroll
    for (int j = 0; j < 4; ++j) {
      const int tt = t0 + 4 * j;
      v2f bmat;
      bmat.x = p[0];        // inputs[b, tt + 2*half,     c0+n]
      bmat.y = p[CD_];      // inputs[b, tt + 2*half + 1, c0+n]
      // 8-byte aligned LDS load of (a[tt+2*half], a[tt+2*half+1])
      v2f amat = *(const v2f*)&sa[tt + 2 * half];
      acc = __builtin_amdgcn_wmma_f32_16x16x4_f32(
          /*neg_a=*/false, amat, /*neg_b=*/false, bmat,
          /*c_mod=*/(short)0, acc, /*reuse_a=*/false, /*reuse_b=*/false);
      p += 4 * CD_;
    }
  }

  // D row M=0 lives in VGPR0 of lanes 0..15 (N = lane).
  if (lane < 16) out_ctx[b * CD_ + c0 + n] = acc[0];
}

// ---------------------------------------------------------------------------
// Phase 3: u_new = sigmoid([ctx, query] @ W_u + b_u)
//          sq_new = sigmoid([ctx, query] @ W_sq + b_sq) + 1
// One block per batch; two fused 1536-dim dot products.
// ---------------------------------------------------------------------------
__global__ __launch_bounds__(256) void gates_kernel(
    const float* __restrict__ ctx, const float* __restrict__ query,
    const float* __restrict__ W_u, const float* __restrict__ b_u,
    const float* __restrict__ W_sq, const float* __restrict__ b_sq,
    float* __restrict__ u_out, float* __restrict__ sq_out) {
  const int b = blockIdx.x;
  float su = 0.0f, ss = 0.0f;
  for (int i = threadIdx.x; i < CD_ + QD_; i += 256) {
    const float x = (i < CD_) ? ctx[b * CD_ + i] : query[b * QD_ + (i - CD_)];
    su += x * W_u[i];
    ss += x * W_sq[i];
  }
  __shared__ float r0[256], r1[256];
  r0[threadIdx.x] = su;
  r1[threadIdx.x] = ss;
  __syncthreads();
  for (int s = 128; s > 0; s >>= 1) {
    if (threadIdx.x < s) {
      r0[threadIdx.x] += r0[threadIdx.x + s];
      r1[threadIdx.x] += r1[threadIdx.x + s];
    }
    __syncthreads();
  }
  if (threadIdx.x == 0) {
    const float xu = r0[0] + b_u[0];
    const float xs = r1[0] + b_sq[0];
    u_out[b]  = 1.0f / (1.0f + __expf(-xu));
    sq_out[b] = 1.0f / (1.0f + __expf(-xs)) + 1.0f;
  }
}

// ---------------------------------------------------------------------------
extern "C" void kernel_launch(void* const* d_in, const int* in_sizes, int n_in,
                              void* d_out, int out_size, void* d_ws, size_t ws_size,
                              hipStream_t stream) {
  (void)in_sizes; (void)n_in; (void)out_size; (void)d_ws; (void)ws_size;
  const float* query  = (const float*)d_in[0];
  const float* inputs = (const float*)d_in[1];
  const float* alpha  = (const float*)d_in[2];
  const float* u      = (const float*)d_in[3];
  const float* sq     = (const float*)d_in[4];
  const float* W_u    = (const float*)d_in[5];
  const float* b_u    = (const float*)d_in[6];
  const float* W_sq   = (const float*)d_in[7];
  const float* b_sq   = (const float*)d_in[8];

  float* out      = (float*)d_out;
  float* out_ctx  = out;                    // [B, CD]
  float* out_a    = out_ctx + B_ * CD_;     // [B, T]
  float* out_u    = out_a   + B_ * T_;      // [B, 1]
  float* out_sq   = out_u   + B_;           // [B, 1]

  alpha_kernel<<<B_, 256, 0, stream>>>(alpha, u, sq, out_a);
  context_kernel<<<(B_ * (CD_ / 16)) / 8, 256, 0, stream>>>(inputs, out_a, out_ctx);
  gates_kernel<<<B_, 256, 0, stream>>>(out_ctx, query, W_u, b_u, W_sq, b_sq,
                                       out_u, out_sq);
}